// Generator_66743791780078
// MI455X (gfx1250) — compile-verified
//
#include <hip/hip_runtime.h>
#include <hip/hip_bf16.h>

typedef float v2f __attribute__((ext_vector_type(2)));
typedef float v8f __attribute__((ext_vector_type(8)));

#define DIMC 128
#define BATCH 8
#define N1 2048
#define N2 4096
#define EPSF 1e-5f

// ---------------------------------------------------------------- KNN
// One thread per query point; refs staged through LDS in tiles of 256.
// Keeps KKEEP smallest (dist, idx) sorted ascending; strict '<' replicates
// jax.lax.top_k's lowest-index-first tie-break.
template <int KKEEP>
__global__ __launch_bounds__(256) void pu_knn_kernel(
    const float* __restrict__ ref, const float* __restrict__ query,
    int Nref, int Nq, int drop_first, int* __restrict__ out_idx) {
  const int b = blockIdx.y;
  const int q = blockIdx.x * 256 + threadIdx.x;
  float qx = 0.f, qy = 0.f, qz = 0.f;
  if (q < Nq) {
    qx = query[((long)b * 3 + 0) * Nq + q];
    qy = query[((long)b * 3 + 1) * Nq + q];
    qz = query[((long)b * 3 + 2) * Nq + q];
  }
  float bd[KKEEP];
  int bi[KKEEP];
#pragma unroll
  for (int t = 0; t < KKEEP; ++t) { bd[t] = 3.4e38f; bi[t] = 0; }

  __shared__ float sx[256], sy[256], sz[256];
  for (int tile = 0; tile < Nref; tile += 256) {
    int r = tile + threadIdx.x;
    if (r < Nref) {
      sx[threadIdx.x] = ref[((long)b * 3 + 0) * Nref + r];
      sy[threadIdx.x] = ref[((long)b * 3 + 1) * Nref + r];
      sz[threadIdx.x] = ref[((long)b * 3 + 2) * Nref + r];
    }
    __syncthreads();
    const int lim = min(256, Nref - tile);
    if (q < Nq) {
      for (int j = 0; j < lim; ++j) {
        float dx = qx - sx[j], dy = qy - sy[j], dz = qz - sz[j];
        float d = dx * dx + dy * dy + dz * dz;
        if (d < bd[KKEEP - 1]) {
          int pos = KKEEP - 1;
#pragma unroll
          for (int t = KKEEP - 2; t >= 0; --t) {
            if (d < bd[t]) { bd[t + 1] = bd[t]; bi[t + 1] = bi[t]; pos = t; }
          }
          bd[pos] = d; bi[pos] = tile + j;
        }
      }
    }
    __syncthreads();
  }
  if (q < Nq) {
#pragma unroll
    for (int t = 0; t < 8; ++t)
      out_idx[((long)b * Nq + q) * 8 + t] = bi[t + drop_first];
  }
}

// ---------------------------------------------------------------- conv0
// out[b,c,k,n] = sum_d W0[c,d]*(xyz[b,d,idx[b,n,k]] - xyz[b,d,n]) + b0[c]
__global__ __launch_bounds__(256) void pu_conv0_kernel(
    const float* __restrict__ xyz, const int* __restrict__ idx,
    const float* __restrict__ W0, const float* __restrict__ b0,
    float* __restrict__ out, int N) {
  __shared__ float sw[DIMC * 3];
  __shared__ float sb[DIMC];
  for (int i = threadIdx.x; i < DIMC * 3; i += 256) sw[i] = W0[i];
  for (int i = threadIdx.x; i < DIMC; i += 256) sb[i] = b0[i];
  __syncthreads();
  long t = (long)blockIdx.x * 256 + threadIdx.x;  // over B*8*N
  if (t >= (long)BATCH * 8 * N) return;
  int n = (int)(t % N);
  int k = (int)((t / N) % 8);
  int b = (int)(t / ((long)8 * N));
  int j = idx[((long)b * N + n) * 8 + k];
  float gx = xyz[((long)b * 3 + 0) * N + j] - xyz[((long)b * 3 + 0) * N + n];
  float gy = xyz[((long)b * 3 + 1) * N + j] - xyz[((long)b * 3 + 1) * N + n];
  float gz = xyz[((long)b * 3 + 2) * N + j] - xyz[((long)b * 3 + 2) * N + n];
  for (int c = 0; c < DIMC; ++c) {
    float v = sw[c * 3 + 0] * gx + sw[c * 3 + 1] * gy + sw[c * 3 + 2] * gz + sb[c];
    out[(((long)b * DIMC + c) * 8 + k) * N + n] = v;
  }
}

// ---------------------------------------------------------------- batch-norm
// x layout (B, C, L); one block per channel
__global__ __launch_bounds__(256) void pu_bn_stats_kernel(
    const float* __restrict__ x, float* __restrict__ stats, int Bn, int L) {
  const int c = blockIdx.x;
  float s = 0.f, s2 = 0.f;
  for (int b = 0; b < Bn; ++b) {
    const float* p = x + ((long)b * DIMC + c) * L;
    for (int i = threadIdx.x; i < L; i += 256) { float v = p[i]; s += v; s2 += v * v; }
  }
  __shared__ float rs[256], rq[256];
  rs[threadIdx.x] = s; rq[threadIdx.x] = s2;
  __syncthreads();
  for (int st = 128; st > 0; st >>= 1) {
    if (threadIdx.x < st) { rs[threadIdx.x] += rs[threadIdx.x + st]; rq[threadIdx.x] += rq[threadIdx.x + st]; }
    __syncthreads();
  }
  if (threadIdx.x == 0) {
    float M = (float)Bn * (float)L;
    float mean = rs[0] / M;
    float var = rq[0] / M - mean * mean;
    stats[c] = mean;
    stats[DIMC + c] = rsqrtf(var + EPSF);
  }
}

__global__ __launch_bounds__(256) void pu_bn_apply_relu_kernel(
    const float* __restrict__ x, float* __restrict__ y,
    const float* __restrict__ stats, const float* __restrict__ gamma,
    const float* __restrict__ beta, long L, long total) {
  long i = (long)blockIdx.x * 256 + threadIdx.x;
  if (i >= total) return;
  int c = (int)((i / L) % DIMC);
  float v = (x[i] - stats[c]) * stats[DIMC + c] * gamma[c] + beta[c];
  y[i] = fmaxf(v, 0.f);
}

// ---------------------------------------------------------------- WMMA GEMM
// Y[b, m, n] = sum_k W[m,k] * X[b,k,n]  (+ bias[m]) (+ shortcut[b,m,n])
// W: 128x128 row-major. X,Y: (B, 128, Ncols).
// One wave computes a 16x64 output tile (4 fp32 accumulators); the A
// fragment (W 16x4 slab) is loaded once per k-step and reused across the
// four 16x16 WMMAs -> 4x matrix-pipe work per unit of address math / L2.
template <bool HAS_BIAS, bool HAS_SHORT>
__global__ __launch_bounds__(256) void pu_wmma_gemm128_kernel(
    const float* __restrict__ W, const float* __restrict__ X,
    float* __restrict__ Y, const float* __restrict__ bias,
    const float* __restrict__ shortcut, int Ncols) {
  const int wave = threadIdx.x >> 5;
  const int lane = threadIdx.x & 31;
  const int ntn = Ncols >> 6;         // 16x64 tiles along N
  const int tilesPerBatch = 8 * ntn;  // (128/16) * (Ncols/64)
  const int tile = blockIdx.x * 8 + wave;
  if (tile >= tilesPerBatch) return;  // wave-uniform
  const int mt = tile / ntn;
  const int nt = tile - mt * ntn;
  const int b = blockIdx.y;
  const int m0 = mt * 16, n0 = nt * 64;
  const int lm = lane & 15;  // column / row-within-half
  const int lh = lane >> 4;  // half select

  const float* Xb = X + (long)b * DIMC * Ncols;
  const float* wrow = W + (long)(m0 + lm) * DIMC + 2 * lh;  // A: K pair per half
  const float* xcol = Xb + n0 + lm;

  v8f acc[4];
#pragma unroll
  for (int t = 0; t < 4; ++t) acc[t] = (v8f){};

#if __has_builtin(__builtin_amdgcn_wmma_f32_16x16x4_f32)
#pragma unroll 4
  for (int k0 = 0; k0 < DIMC; k0 += 4) {
    v2f a = *(const v2f*)(wrow + k0);  // A[m0+lm][k0+2*lh .. +1]
    long r = (long)(k0 + 2 * lh) * Ncols;
#pragma unroll
    for (int t = 0; t < 4; ++t) {
      v2f bb;
      bb.x = xcol[r + 16 * t];          // B[k0+2*lh  ][n0+16t+lm]
      bb.y = xcol[r + Ncols + 16 * t];  // B[k0+2*lh+1][n0+16t+lm]
      acc[t] = __builtin_amdgcn_wmma_f32_16x16x4_f32(
          false, a, false, bb, (short)0, acc[t], false, false);
    }
  }
#else
#pragma unroll
  for (int t = 0; t < 4; ++t)
    for (int r = 0; r < 8; ++r) {
      int row = m0 + lh * 8 + r;
      float v = 0.f;
      for (int k = 0; k < DIMC; ++k)
        v += W[(long)row * DIMC + k] * Xb[(long)k * Ncols + n0 + 16 * t + lm];
      acc[t][r] = v;
    }
#endif

  float bv[8];
  if (HAS_BIAS) {
    // bias[m0+lh*8 .. +7]: two aligned float4 register loads, no branches
    const float4 b0 = *(const float4*)(bias + m0 + lh * 8);
    const float4 b1 = *(const float4*)(bias + m0 + lh * 8 + 4);
    bv[0] = b0.x; bv[1] = b0.y; bv[2] = b0.z; bv[3] = b0.w;
    bv[4] = b1.x; bv[5] = b1.y; bv[6] = b1.z; bv[7] = b1.w;
  }
  float* Yb = Y + (long)b * DIMC * Ncols;
  const float* Sb = HAS_SHORT ? (shortcut + (long)b * DIMC * Ncols) : nullptr;
#pragma unroll
  for (int r = 0; r < 8; ++r) {
    const int row = m0 + lh * 8 + r;  // C/D layout: vgpr r, +8 for upper half-lanes
    const long o0 = (long)row * Ncols + n0 + lm;
#pragma unroll
    for (int t = 0; t < 4; ++t) {
      float v = acc[t][r];
      if (HAS_BIAS) v += bv[r];
      if (HAS_SHORT) v += Sb[o0 + 16 * t];
      Yb[o0 + 16 * t] = v;
    }
  }
}

// ---------------------------------------------------------------- aggregate
// agg[b,c,n] = (p[b,c,n] + sum_k p[b,c,idx[b,n,k]]) / 9
__global__ __launch_bounds__(256) void pu_aggregate_kernel(
    const float* __restrict__ p, const int* __restrict__ idx,
    float* __restrict__ agg, int N) {
  int n = blockIdx.x * 256 + threadIdx.x;
  int b = blockIdx.y;
  if (n >= N) return;
  int j[8];
#pragma unroll
  for (int t = 0; t < 8; ++t) j[t] = idx[((long)b * N + n) * 8 + t];
  const float* pb = p + (long)b * DIMC * N;
  float* ab = agg + (long)b * DIMC * N;
  const float inv9 = 1.0f / 9.0f;
  for (int c = 0; c < DIMC; ++c) {
    const float* pc = pb + (long)c * N;
    float s = pc[n];
#pragma unroll
    for (int t = 0; t < 8; ++t) s += pc[j[t]];
    ab[(long)c * N + n] = s * inv9;
  }
}

// ---------------------------------------------------------------- new_xyz
// out6 = (Wc@p_center + bc + Wn@(sum_nbr p) + 8*bn)/9 ; reshape(3,UP)+xyz
__global__ __launch_bounds__(256) void pu_new_xyz_kernel(
    const float* __restrict__ p, const int* __restrict__ idx,
    const float* __restrict__ xyz, const float* __restrict__ Wc,
    const float* __restrict__ bc, const float* __restrict__ Wn,
    const float* __restrict__ bnn, float* __restrict__ out, int N) {
  __shared__ float swc[6 * DIMC], swn[6 * DIMC], sbc[6], sbn[6];
  for (int i = threadIdx.x; i < 6 * DIMC; i += 256) { swc[i] = Wc[i]; swn[i] = Wn[i]; }
  if (threadIdx.x < 6) { sbc[threadIdx.x] = bc[threadIdx.x]; sbn[threadIdx.x] = bnn[threadIdx.x]; }
  __syncthreads();
  int n = blockIdx.x * 256 + threadIdx.x;
  int b = blockIdx.y;
  if (n >= N) return;
  int j[8];
#pragma unroll
  for (int t = 0; t < 8; ++t) j[t] = idx[((long)b * N + n) * 8 + t];
  const float* pb = p + (long)b * DIMC * N;
  float acc[6];
#pragma unroll
  for (int o = 0; o < 6; ++o) acc[o] = sbc[o] + 8.f * sbn[o];
  for (int c = 0; c < DIMC; ++c) {
    const float* pc = pb + (long)c * N;
    float ctr = pc[n];
    float s = 0.f;
#pragma unroll
    for (int t = 0; t < 8; ++t) s += pc[j[t]];
#pragma unroll
    for (int o = 0; o < 6; ++o) acc[o] += swc[o * DIMC + c] * ctr + swn[o * DIMC + c] * s;
  }
  const float inv9 = 1.0f / 9.0f;
#pragma unroll
  for (int d = 0; d < 3; ++d)
#pragma unroll
    for (int u = 0; u < 2; ++u)
      out[((long)b * 3 + d) * (2 * N) + (long)u * N + n] =
          acc[d * 2 + u] * inv9 + xyz[((long)b * 3 + d) * N + n];
}

// ---------------------------------------------------------------- gather mean
__global__ __launch_bounds__(256) void pu_gather_mean_kernel(
    const float* __restrict__ pin, const int* __restrict__ idx,
    float* __restrict__ pout, int Nref, int Nq) {
  int m = blockIdx.x * 256 + threadIdx.x;
  int b = blockIdx.y;
  if (m >= Nq) return;
  int j[8];
#pragma unroll
  for (int t = 0; t < 8; ++t) j[t] = idx[((long)b * Nq + m) * 8 + t];
  for (int c = 0; c < DIMC; ++c) {
    const float* pc = pin + ((long)b * DIMC + c) * Nref;
    float s = 0.f;
#pragma unroll
    for (int t = 0; t < 8; ++t) s += pc[j[t]];
    pout[((long)b * DIMC + c) * Nq + m] = s * 0.125f;
  }
}

// ---------------------------------------------------------------- max over K
__global__ __launch_bounds__(256) void pu_max_k_kernel(
    const float* __restrict__ x, float* __restrict__ y, int N) {
  long i = (long)blockIdx.x * 256 + threadIdx.x;  // over B*C*N
  if (i >= (long)BATCH * DIMC * N) return;
  int n = (int)(i % N);
  long bc = i / N;
  const float* p = x + (bc * 8) * (long)N + n;
  float m = p[0];
#pragma unroll
  for (int k = 1; k < 8; ++k) m = fmaxf(m, p[(long)k * N]);
  y[bc * (long)N + n] = m;
}

// ================================================================ host
extern "C" void kernel_launch(void* const* d_in, const int* in_sizes, int n_in,
                              void* d_out, int out_size, void* d_ws, size_t ws_size,
                              hipStream_t stream) {
  const float* xyz    = (const float*)d_in[0];   // (8,3,2048)
  const float* fn_W0  = (const float*)d_in[1];   // (128,3)
  const float* fn_b0  = (const float*)d_in[2];   // (128)
  const float* fn_W   = (const float*)d_in[3];   // (2,128,128)
  const float* fn_b   = (const float*)d_in[4];   // (2,128)
  const float* fn_bng = (const float*)d_in[5];   // (3,128)
  const float* fn_bnb = (const float*)d_in[6];   // (3,128)
  const float* rb_bng = (const float*)d_in[7];   // (2,12,128)
  const float* rb_bnb = (const float*)d_in[8];
  const float* rb_W   = (const float*)d_in[9];   // (2,12,128,128)
  const float* rb_b   = (const float*)d_in[10];  // (2,12,128)
  const float* rb_cW  = (const float*)d_in[11];  // (2,6,128)
  const float* rb_cb  = (const float*)d_in[12];  // (2,6)
  const float* rb_nW  = (const float*)d_in[13];  // (2,6,128)
  const float* rb_nb  = (const float*)d_in[14];  // (2,6)
  float* outp = (float*)d_out;                   // (8,3,8192)

  float* ws = (float*)d_ws;
  size_t off = 0;
  int* idx1 = (int*)(ws + off); off += (size_t)BATCH * N1 * 8;
  int* idx2 = (int*)(ws + off); off += (size_t)BATCH * N2 * 8;
  int* idxI = (int*)(ws + off); off += (size_t)BATCH * N2 * 8;
  float* stats   = ws + off; off += 2 * DIMC;
  float* points1 = ws + off; off += (size_t)BATCH * DIMC * N1;
  float* xyz2    = ws + off; off += (size_t)BATCH * 3 * N2;
  float* bufA    = ws + off; off += (size_t)BATCH * DIMC * 8 * N1;
  float* bufB    = ws + off;  // also 8*128*8*2048; reused after featurenet:
  float* points2 = bufB;
  float* pbuf    = bufB + (size_t)BATCH * DIMC * N2;
  float* agg     = bufB + 2 * (size_t)BATCH * DIMC * N2;

  const long FL = (long)8 * N1;             // featurenet per-(b,c) length
  const long ftot = (long)BATCH * DIMC * FL;

  // ---- featurenet ----
  pu_knn_kernel<9><<<dim3(N1 / 256, BATCH), 256, 0, stream>>>(xyz, xyz, N1, N1, 1, idx1);
  pu_conv0_kernel<<<(BATCH * 8 * N1) / 256, 256, 0, stream>>>(xyz, idx1, fn_W0, fn_b0, bufA, N1);
  pu_bn_stats_kernel<<<DIMC, 256, 0, stream>>>(bufA, stats, BATCH, (int)FL);
  pu_bn_apply_relu_kernel<<<(ftot + 255) / 256, 256, 0, stream>>>(bufA, bufA, stats, fn_bng, fn_bnb, FL, ftot);
  for (int l = 0; l < 2; ++l) {
    const float* src = (l == 0) ? bufA : bufB;
    float* dst = (l == 0) ? bufB : bufA;
    int ncols = 8 * N1;  // 16384
    pu_wmma_gemm128_kernel<true, false>
        <<<dim3((8 * (ncols / 64)) / 8, BATCH), 256, 0, stream>>>(
            fn_W + (size_t)l * DIMC * DIMC, src, dst, fn_b + (size_t)l * DIMC, nullptr, ncols);
    pu_bn_stats_kernel<<<DIMC, 256, 0, stream>>>(dst, stats, BATCH, (int)FL);
    pu_bn_apply_relu_kernel<<<(ftot + 255) / 256, 256, 0, stream>>>(
        dst, dst, stats, fn_bng + (size_t)(l + 1) * DIMC, fn_bnb + (size_t)(l + 1) * DIMC, FL, ftot);
  }
  pu_max_k_kernel<<<((long)BATCH * DIMC * N1 + 255) / 256, 256, 0, stream>>>(bufA, points1, N1);

  // knn for stage 2 happens after xyz2 exists; stage-1 idx == idx1 (same xyz)
  for (int j = 0; j < 2; ++j) {
    const int N = (j == 0) ? N1 : N2;
    const float* xyz_s = (j == 0) ? xyz : xyz2;
    float* points = (j == 0) ? points1 : points2;
    int* idx = (j == 0) ? idx1 : idx2;
    float* out_xyz = (j == 0) ? xyz2 : outp;
    if (j == 1)
      pu_knn_kernel<9><<<dim3(N2 / 256, BATCH), 256, 0, stream>>>(xyz2, xyz2, N2, N2, 1, idx2);
    const long tot = (long)BATCH * DIMC * N;
    for (int i = 0; i < 12; ++i) {
      const size_t li = (size_t)j * 12 + i;
      pu_bn_stats_kernel<<<DIMC, 256, 0, stream>>>(points, stats, BATCH, N);
      pu_bn_apply_relu_kernel<<<(tot + 255) / 256, 256, 0, stream>>>(
          points, pbuf, stats, rb_bng + li * DIMC, rb_bnb + li * DIMC, N, tot);
      pu_aggregate_kernel<<<dim3(N / 256, BATCH), 256, 0, stream>>>(pbuf, idx, agg, N);
      if (i == 11)
        pu_new_xyz_kernel<<<dim3(N / 256, BATCH), 256, 0, stream>>>(
            pbuf, idx, xyz_s, rb_cW + (size_t)j * 6 * DIMC, rb_cb + (size_t)j * 6,
            rb_nW + (size_t)j * 6 * DIMC, rb_nb + (size_t)j * 6, out_xyz, N);
      // points = W@agg + b + points (in-place shortcut; each element RMW once)
      pu_wmma_gemm128_kernel<true, true>
          <<<dim3((8 * (N / 64)) / 8, BATCH), 256, 0, stream>>>(
              rb_W + li * DIMC * DIMC, agg, points, rb_b + li * DIMC, points, N);
    }
    if (j == 0) {
      pu_knn_kernel<8><<<dim3(N2 / 256, BATCH), 256, 0, stream>>>(xyz, xyz2, N1, N2, 0, idxI);
      pu_gather_mean_kernel<<<dim3(N2 / 256, BATCH), 256, 0, stream>>>(points1, idxI, points2, N1, N2);
    }
  }
}